// MoEGRU_31284541784554
// MI455X (gfx1250) — compile-verified
//
#include <hip/hip_runtime.h>
#include <hip/hip_bf16.h>

// Problem constants (match reference)
#define B_   256
#define L_   1800
#define F_   50
#define E_   4
#define H_   32
#define D_   64
#define G_   96     // 3*H
#define HU_  32
#define PCH  24     // proj chunk; 1800 % 24 == 0

typedef __attribute__((ext_vector_type(2))) float v2f;
typedef __attribute__((ext_vector_type(8))) float v8f;

__device__ __forceinline__ v8f wmma4(v2f a, v2f b, v8f c) {
  // D = A(16x4, f32) * B(4x16, f32) + C(16x16, f32)
  return __builtin_amdgcn_wmma_f32_16x16x4_f32(false, a, false, b, (short)0, c,
                                               false, false);
}

__device__ __forceinline__ float sigm_(float x) {
  return 1.f / (1.f + __expf(-x));
}
__device__ __forceinline__ float tanh_(float x) {
  x = fminf(fmaxf(x, -15.f), 15.f);
  float t = __expf(2.f * x);
  return (t - 1.f) / (t + 1.f);
}

// -------- Phase 1: x_proj[l][b][d] = x[b,l,:]·W_in[d,:] + b_in[d] + emb[horizon[b]][d]
__global__ __launch_bounds__(64)
void proj_kernel(const float* __restrict__ x, const int* __restrict__ horizon,
                 const float* __restrict__ W_in, const float* __restrict__ b_in,
                 const float* __restrict__ emb, float* __restrict__ xp) {
  const int blk = blockIdx.x;
  const int b   = blk / (L_ / PCH);
  const int l0  = (blk % (L_ / PCH)) * PCH;
  const int t   = threadIdx.x;   // d index, 0..63

  float w[F_];
#pragma unroll
  for (int f = 0; f < F_; ++f) w[f] = W_in[t * F_ + f];
  const float base = b_in[t] + emb[horizon[b] * D_ + t];

  __shared__ float xs[PCH][F_ + 2];
  const float* xrow = x + ((size_t)b * L_ + l0) * F_;
  for (int i = t; i < PCH * F_; i += 64) xs[i / F_][i % F_] = xrow[i];
  __syncthreads();

  for (int ll = 0; ll < PCH; ++ll) {
    float acc = base;
#pragma unroll
    for (int f = 0; f < F_; ++f) acc += xs[ll][f] * w[f];
    xp[((size_t)(l0 + ll) * B_ + b) * D_ + t] = acc;
  }
}

// -------- Phase 2: gating (top-2 softmax, dense routing weights)
__global__ void gate_kernel(const int* __restrict__ horizon,
                            const float* __restrict__ emb,
                            const float* __restrict__ W_gate,
                            const float* __restrict__ b_gate,
                            float* __restrict__ wdense) {
  int b = threadIdx.x;
  if (b >= B_) return;
  const float* he = emb + horizon[b] * D_;
  float logits[E_];
  for (int e = 0; e < E_; ++e) {
    float a = b_gate[e];
    for (int d = 0; d < D_; ++d) a += he[d] * W_gate[e * D_ + d];
    logits[e] = a;
  }
  // top-2, ties broken by lowest index (JAX top_k semantics)
  int i0 = 0;
  for (int e = 1; e < E_; ++e) if (logits[e] > logits[i0]) i0 = e;
  int i1 = -1;
  for (int e = 0; e < E_; ++e) {
    if (e == i0) continue;
    if (i1 < 0 || logits[e] > logits[i1]) i1 = e;
  }
  float m  = logits[i0];
  float e0 = __expf(logits[i0] - m), e1 = __expf(logits[i1] - m);
  float s  = e0 + e1;
  for (int e = 0; e < E_; ++e) wdense[b * E_ + e] = 0.f;
  wdense[b * E_ + i0] = e0 / s;
  wdense[b * E_ + i1] = e1 / s;
}

// -------- Phase 3: per-(expert, 16-batch-tile) fused 2-layer GRU scan + head MLP.
// One wave (32 lanes) per workgroup. All GEMMs via V_WMMA_F32_16X16X4_F32,
// gate-major: C[M=gate][N=batch]. Lane n<16 holds rows i+0..7 of each tile for
// batch n; lane n+16 holds rows 8..15 for batch n. Hidden state kept in LDS as
// h[batch][unit] which is exactly the B-operand access pattern (b64 per chunk).
__global__ __launch_bounds__(32)
void expert_kernel(const float* __restrict__ xp,
                   const float* __restrict__ Wih0, const float* __restrict__ Whh0,
                   const float* __restrict__ bih0, const float* __restrict__ bhh0,
                   const float* __restrict__ Wih1, const float* __restrict__ Whh1,
                   const float* __restrict__ bih1, const float* __restrict__ bhh1,
                   const float* __restrict__ Wh1,  const float* __restrict__ bh1,
                   const float* __restrict__ Wh2,  const float* __restrict__ bh2,
                   float* __restrict__ preds) {
  const int e  = blockIdx.x >> 4;     // expert
  const int bt = blockIdx.x & 15;     // batch tile
  const int b0 = bt * 16;

  __shared__ float sWih0[G_ * D_];    // 24576 B
  __shared__ float sWhh0[G_ * H_];    // 12288 B
  __shared__ float sWih1[G_ * H_];    // 12288 B
  __shared__ float sWhh1[G_ * H_];    // 12288 B
  __shared__ float sBrz0[2 * H_], sBxn0[H_], sBhn0[H_];
  __shared__ float sBrz1[2 * H_], sBxn1[H_], sBhn1[H_];
  __shared__ float h0[16 * H_];       // [batch][unit]
  __shared__ float h1[16 * H_];

  const int lane = threadIdx.x;
  for (int i = lane; i < G_ * D_; i += 32) sWih0[i] = Wih0[e * G_ * D_ + i];
  for (int i = lane; i < G_ * H_; i += 32) {
    sWhh0[i] = Whh0[e * G_ * H_ + i];
    sWih1[i] = Wih1[e * G_ * H_ + i];
    sWhh1[i] = Whh1[e * G_ * H_ + i];
  }
  for (int i = lane; i < 2 * H_; i += 32) {
    sBrz0[i] = bih0[e * G_ + i] + bhh0[e * G_ + i];
    sBrz1[i] = bih1[e * G_ + i] + bhh1[e * G_ + i];
  }
  for (int i = lane; i < H_; i += 32) {
    sBxn0[i] = bih0[e * G_ + 2 * H_ + i];
    sBhn0[i] = bhh0[e * G_ + 2 * H_ + i];
    sBxn1[i] = bih1[e * G_ + 2 * H_ + i];
    sBhn1[i] = bhh1[e * G_ + 2 * H_ + i];
  }
  for (int i = lane; i < 16 * H_; i += 32) { h0[i] = 0.f; h1[i] = 0.f; }
  __syncthreads();

  const int lm   = lane & 15;    // batch within tile (= N), also M row offset base
  const int hi   = lane >> 4;    // half: selects K pair (A/B) and M+8 (C)
  const int koff = 2 * hi;

  const v8f vzero = {};

  for (int l = 0; l < L_; ++l) {
    // ============ layer 0 ============
    v8f acc[4], xn[2], hn[2];
#pragma unroll
    for (int t = 0; t < 4; ++t) acc[t] = vzero;
#pragma unroll
    for (int t = 0; t < 2; ++t) { xn[t] = vzero; hn[t] = vzero; }

    const float* xrow = xp + ((size_t)l * B_ + b0 + lm) * D_;
    // x-gates: K = D = 64 (16 chunks). B operand straight from global (cached).
    for (int kc = 0; kc < 16; ++kc) {
      const int k = kc * 4 + koff;
      v2f bx; bx.x = xrow[k]; bx.y = xrow[k + 1];
#pragma unroll
      for (int t = 0; t < 4; ++t) {
        const int g = t * 16 + lm;
        v2f a; a.x = sWih0[g * D_ + k]; a.y = sWih0[g * D_ + k + 1];
        acc[t] = wmma4(a, bx, acc[t]);
      }
#pragma unroll
      for (int t = 0; t < 2; ++t) {
        const int g = (4 + t) * 16 + lm;
        v2f a; a.x = sWih0[g * D_ + k]; a.y = sWih0[g * D_ + k + 1];
        xn[t] = wmma4(a, bx, xn[t]);
      }
    }
    // h-gates: K = H = 32 (8 chunks). B operand = h0[batch][unit] from LDS.
    for (int kc = 0; kc < 8; ++kc) {
      const int k = kc * 4 + koff;
      v2f bh; bh.x = h0[lm * H_ + k]; bh.y = h0[lm * H_ + k + 1];
#pragma unroll
      for (int t = 0; t < 4; ++t) {
        const int g = t * 16 + lm;
        v2f a; a.x = sWhh0[g * H_ + k]; a.y = sWhh0[g * H_ + k + 1];
        acc[t] = wmma4(a, bh, acc[t]);
      }
#pragma unroll
      for (int t = 0; t < 2; ++t) {
        const int g = (4 + t) * 16 + lm;
        v2f a; a.x = sWhh0[g * H_ + k]; a.y = sWhh0[g * H_ + k + 1];
        hn[t] = wmma4(a, bh, hn[t]);
      }
    }
    // elementwise GRU cell, lane-local: lane holds all gates of batch lm for
    // units u = t*16 + 8*hi + i.
#pragma unroll
    for (int t = 0; t < 2; ++t) {
#pragma unroll
      for (int i = 0; i < 8; ++i) {
        const int u = t * 16 + 8 * hi + i;
        float r  = sigm_(acc[t][i]     + sBrz0[u]);
        float z  = sigm_(acc[t + 2][i] + sBrz0[H_ + u]);
        float nn = tanh_(xn[t][i] + sBxn0[u] + r * (hn[t][i] + sBhn0[u]));
        float hp = h0[lm * H_ + u];
        h0[lm * H_ + u] = (1.f - z) * nn + z * hp;
      }
    }
    __syncthreads();   // single-wave WG: S_NOP in HW; orders LDS for compiler

    // ============ layer 1 ============
#pragma unroll
    for (int t = 0; t < 4; ++t) acc[t] = vzero;
#pragma unroll
    for (int t = 0; t < 2; ++t) { xn[t] = vzero; hn[t] = vzero; }
    for (int kc = 0; kc < 8; ++kc) {
      const int k = kc * 4 + koff;
      v2f bx; bx.x = h0[lm * H_ + k]; bx.y = h0[lm * H_ + k + 1];
      v2f bh; bh.x = h1[lm * H_ + k]; bh.y = h1[lm * H_ + k + 1];
#pragma unroll
      for (int t = 0; t < 4; ++t) {
        const int g = t * 16 + lm;
        v2f a1; a1.x = sWih1[g * H_ + k]; a1.y = sWih1[g * H_ + k + 1];
        acc[t] = wmma4(a1, bx, acc[t]);
        v2f a2; a2.x = sWhh1[g * H_ + k]; a2.y = sWhh1[g * H_ + k + 1];
        acc[t] = wmma4(a2, bh, acc[t]);
      }
#pragma unroll
      for (int t = 0; t < 2; ++t) {
        const int g = (4 + t) * 16 + lm;
        v2f a1; a1.x = sWih1[g * H_ + k]; a1.y = sWih1[g * H_ + k + 1];
        xn[t] = wmma4(a1, bx, xn[t]);
        v2f a2; a2.x = sWhh1[g * H_ + k]; a2.y = sWhh1[g * H_ + k + 1];
        hn[t] = wmma4(a2, bh, hn[t]);
      }
    }
#pragma unroll
    for (int t = 0; t < 2; ++t) {
#pragma unroll
      for (int i = 0; i < 8; ++i) {
        const int u = t * 16 + 8 * hi + i;
        float r  = sigm_(acc[t][i]     + sBrz1[u]);
        float z  = sigm_(acc[t + 2][i] + sBrz1[H_ + u]);
        float nn = tanh_(xn[t][i] + sBxn1[u] + r * (hn[t][i] + sBhn1[u]));
        float hp = h1[lm * H_ + u];
        h1[lm * H_ + u] = (1.f - z) * nn + z * hp;
      }
    }
    __syncthreads();
  }

  // ============ head MLP (tiny, scalar) ============
  float* hid = h0;   // reuse LDS (same extent 16*32)
  const float* wh1 = Wh1 + e * HU_ * H_;
  for (int j = 0; j < 16; ++j) {
    const int u = hi * 16 + j;
    float a = bh1[e * HU_ + u];
    for (int k = 0; k < H_; ++k) a += h1[lm * H_ + k] * wh1[u * H_ + k];
    hid[lm * HU_ + u] = fmaxf(a, 0.f);
  }
  __syncthreads();
  if (lane < 16) {
    float a = bh2[e];
    for (int u = 0; u < HU_; ++u) a += hid[lane * HU_ + u] * Wh2[e * HU_ + u];
    preds[e * B_ + b0 + lane] = a;
  }
}

// -------- Phase 4: deterministic combine (no atomics)
__global__ void combine_kernel(const float* __restrict__ wdense,
                               const float* __restrict__ preds,
                               float* __restrict__ out) {
  int b = threadIdx.x;
  if (b >= B_) return;
  float a = 0.f;
  for (int e = 0; e < E_; ++e) a += wdense[b * E_ + e] * preds[e * B_ + b];
  out[b] = a;
}

extern "C" void kernel_launch(void* const* d_in, const int* in_sizes, int n_in,
                              void* d_out, int out_size, void* d_ws, size_t ws_size,
                              hipStream_t stream) {
  (void)in_sizes; (void)n_in; (void)out_size; (void)ws_size;
  const float* x       = (const float*)d_in[0];
  const int*   horizon = (const int*)  d_in[1];
  const float* W_in    = (const float*)d_in[2];
  const float* b_in    = (const float*)d_in[3];
  const float* emb     = (const float*)d_in[4];
  const float* W_gate  = (const float*)d_in[5];
  const float* b_gate  = (const float*)d_in[6];
  const float* Wih0    = (const float*)d_in[7];
  const float* Whh0    = (const float*)d_in[8];
  const float* bih0    = (const float*)d_in[9];
  const float* bhh0    = (const float*)d_in[10];
  const float* Wih1    = (const float*)d_in[11];
  const float* Whh1    = (const float*)d_in[12];
  const float* bih1    = (const float*)d_in[13];
  const float* bhh1    = (const float*)d_in[14];
  const float* Wh1     = (const float*)d_in[15];
  const float* bh1     = (const float*)d_in[16];
  const float* Wh2     = (const float*)d_in[17];
  const float* bh2     = (const float*)d_in[18];

  float* ws     = (float*)d_ws;
  float* xp     = ws;                                  // L*B*D f32 (~118 MB)
  float* wdense = ws + (size_t)L_ * B_ * D_;           // B*E
  float* preds  = wdense + B_ * E_;                    // E*B

  proj_kernel<<<B_ * (L_ / PCH), 64, 0, stream>>>(x, horizon, W_in, b_in, emb, xp);
  gate_kernel<<<1, B_, 0, stream>>>(horizon, emb, W_gate, b_gate, wdense);
  expert_kernel<<<E_ * 16, 32, 0, stream>>>(xp, Wih0, Whh0, bih0, bhh0,
                                            Wih1, Whh1, bih1, bhh1,
                                            Wh1, bh1, Wh2, bh2, preds);
  combine_kernel<<<1, B_, 0, stream>>>(wdense, preds, (float*)d_out);
}